// SkeletonNet_28484223107634
// MI455X (gfx1250) — compile-verified
//
#include <hip/hip_runtime.h>
#include <hip/hip_bf16.h>
#include <math.h>

// ---------------------------------------------------------------------------
// SkeletonNet forward for MI455X (gfx1250, wave32).
//   B=4, N=32768, SKP=512, seeds=2048, K1=32, K2=2, SPHERE=128
// ---------------------------------------------------------------------------

typedef __attribute__((ext_vector_type(2))) float v2f;
typedef __attribute__((ext_vector_type(8))) float v8f;

#define NBATCH   4
#define NPTS     32768
#define NSEED    2048      // SKP_NUM * 4
#define NSKP     512
#define KNN1     32
#define KNN2     2
#define SPHERE_N 128
#define RAD_SCALE 1.5f
#define CHUNK    256       // points staged to LDS per block iteration

// order-preserving float <-> u32 transforms (for packed top-K keys)
__device__ __forceinline__ unsigned key_of(float f) {
  unsigned u = __float_as_uint(f);
  unsigned m = (unsigned)((int)u >> 31) | 0x80000000u;
  return u ^ m;
}
__device__ __forceinline__ float val_of(unsigned k) {
  unsigned m = (k & 0x80000000u) ? 0x80000000u : 0xFFFFFFFFu;
  return __uint_as_float(k ^ m);
}

// ---------------------------------------------------------------------------
// p2[b][n] = |pts[b][n]|^2
// ---------------------------------------------------------------------------
__global__ __launch_bounds__(256)
void p2_kernel(const float* __restrict__ pts, float* __restrict__ p2, int total) {
  int i = blockIdx.x * blockDim.x + threadIdx.x;
  if (i >= total) return;
  float x = pts[3 * (size_t)i + 0];
  float y = pts[3 * (size_t)i + 1];
  float z = pts[3 * (size_t)i + 2];
  p2[i] = x * x + y * y + z * z;
}

// ---------------------------------------------------------------------------
// wave32 argmax reduce (prefer smaller index on ties, matching jnp.argmax)
// ---------------------------------------------------------------------------
__device__ __forceinline__ void wave_argmax(float& m, int& ix) {
#pragma unroll
  for (int off = 16; off > 0; off >>= 1) {
    float om = __shfl_xor(m, off, 32);
    int   oi = __shfl_xor(ix, off, 32);
    if (om > m || (om == m && oi < ix)) { m = om; ix = oi; }
  }
}

// ---------------------------------------------------------------------------
// Furthest point sampling. One block of 1024 threads per batch.
// N = PT*1024 points; selects NS samples (first sample = index 0, as in ref).
// dist[] lives entirely in VGPRs (per-thread fixed partition).
// ---------------------------------------------------------------------------
template <int PT, int NS>
__global__ __launch_bounds__(1024)
void fps_kernel(const float* __restrict__ pts,
                float* __restrict__ cout,
                const float* __restrict__ rad_in,
                float* __restrict__ rad_out) {
  constexpr int N = PT * 1024;
  const int b    = blockIdx.x;
  const int tid  = threadIdx.x;
  const int wv   = tid >> 5;
  const int lane = tid & 31;
  const float* pb = pts + (size_t)b * N * 3;

  __shared__ float s_max[32];
  __shared__ int   s_idx[32];
  __shared__ int   s_best;
  __shared__ int   sids[NS];

  float dst[PT];
#pragma unroll
  for (int q = 0; q < PT; ++q) dst[q] = 1e10f;

  if (tid == 0) sids[0] = 0;
  int last = 0;

  for (int i = 1; i < NS; ++i) {
    const float lx = pb[3 * (size_t)last + 0];
    const float ly = pb[3 * (size_t)last + 1];
    const float lz = pb[3 * (size_t)last + 2];
    float lmax = -1e30f;
    int   lidx = 0x7fffffff;
#pragma unroll
    for (int q = 0; q < PT; ++q) {
      const int j = tid + q * 1024;
      const float* pp = pb + 3 * (size_t)j;
      const float dx = pp[0] - lx, dy = pp[1] - ly, dz = pp[2] - lz;
      const float d  = dx * dx + dy * dy + dz * dz;
      const float dm = fminf(dst[q], d);
      dst[q] = dm;
      if (dm > lmax) { lmax = dm; lidx = j; }
    }
    wave_argmax(lmax, lidx);
    __syncthreads();                    // protect s_max/s_idx reuse
    if (lane == 0) { s_max[wv] = lmax; s_idx[wv] = lidx; }
    __syncthreads();
    if (wv == 0) {
      float m = s_max[lane];
      int  ix = s_idx[lane];
      wave_argmax(m, ix);
      if (lane == 0) { s_best = ix; sids[i] = ix; }
    }
    __syncthreads();
    last = s_best;
  }

  __syncthreads();
  for (int i = tid; i < NS; i += 1024) {
    const int id = sids[i];
    const size_t o = ((size_t)b * NS + i) * 3;
    cout[o + 0] = pb[3 * (size_t)id + 0];
    cout[o + 1] = pb[3 * (size_t)id + 1];
    cout[o + 2] = pb[3 * (size_t)id + 2];
    if (rad_in) rad_out[(size_t)b * NS + i] = rad_in[(size_t)b * N + id];
  }
}

// ---------------------------------------------------------------------------
// Skeletonization: for each of M centers find K nearest of N points,
// new center = mean of neighbors, radius = mean distance.
//
// d2 = c2 + p2 - 2*cp (as in the reference); the cp cross term is a K=3
// (padded to 4) matmul done with V_WMMA_F32_16X16X4_F32.
//
// Block = 256 threads = 8 waves; each wave owns 32 centers (2 WMMA row
// tiles). The block cooperatively stages CHUNK=256 points (+ their |p|^2)
// into LDS with global_load_async_to_lds_b128 (ASYNCcnt), then each wave
// walks the staged chunk in 16-point sub-tiles: 2 WMMAs -> 32x16 cp tile in
// LDS ([32][17] rows -> bank-conflict free) -> per-lane branchless top-K on
// packed u64 (key<<32 | idx) keys. Selection key is s = p2 - 2*cp (c2 is
// lane-constant -> order invariant); packed keys make ties resolve to the
// smaller index, matching stable lax.top_k.
// ---------------------------------------------------------------------------
template <int K, int M, int N>
__global__ __launch_bounds__(256)
void sklnz_kernel(const float* __restrict__ cin,
                  const float* __restrict__ pts,
                  const float* __restrict__ p2,
                  float* __restrict__ cout,
                  float* __restrict__ rad) {
  constexpr int MB = M / 256;                 // blocks per batch
  const int b    = blockIdx.x / MB;
  const int mblk = blockIdx.x % MB;
  const int tid  = threadIdx.x;
  const int wv   = tid >> 5;
  const int l    = tid & 31;
  const int lm   = l & 15;
  const bool lo  = (l < 16);

  __shared__ float lds_cp[8][32][17];
  __shared__ float lds_pts[CHUNK * 3];
  __shared__ float lds_p2[CHUNK];

  const float* cwb = cin + ((size_t)b * M + mblk * 256 + wv * 32) * 3;
  const float* pb  = pts + (size_t)b * N * 3;
  const float* q2  = p2 + (size_t)b * N;

  // A operands (loop invariant). A 16x4 layout: lanes 0-15 -> K=0,1 ;
  // lanes 16-31 -> K=2,3 (K=3 padded with 0).
  const float* a0p = cwb + 3 * (size_t)lm;         // tile0: centers 0..15
  const float* a1p = cwb + 3 * (size_t)(16 + lm);  // tile1: centers 16..31
  v2f A0, A1;
  A0.x = lo ? a0p[0] : a0p[2];
  A0.y = lo ? a0p[1] : 0.0f;
  A1.x = lo ? a1p[0] : a1p[2];
  A1.y = lo ? a1p[1] : 0.0f;

  // this lane's own center (row l of the wave's 32x* tile)
  const float ccx = cwb[3 * (size_t)l + 0];
  const float ccy = cwb[3 * (size_t)l + 1];
  const float ccz = cwb[3 * (size_t)l + 2];
  const float c2own = ccx * ccx + ccy * ccy + ccz * ccz;

  unsigned long long L[K];
#pragma unroll
  for (int q = 0; q < K; ++q) L[q] = 0xFFFFFFFFFFFFFFFFull;

  const v8f zeroC = {0.f, 0.f, 0.f, 0.f, 0.f, 0.f, 0.f, 0.f};
  const int rowhalf = lo ? 0 : 8;

  // LDS byte offsets for the async copies (workgroup-relative)
  const unsigned lds_pts_off = (unsigned)(uintptr_t)(void*)&lds_pts[0];
  const unsigned lds_p2_off  = (unsigned)(uintptr_t)(void*)&lds_p2[0];

#pragma unroll 1
  for (int n0 = 0; n0 < N; n0 += CHUNK) {
    // --- cooperative async stage of CHUNK points + |p|^2 into LDS ---------
    // xyz: CHUNK*3 = 768 dwords = 192 x b128 ; p2: CHUNK = 256 dwords = 64 x b128
    {
      unsigned long long ga;
      unsigned ldso;
      if (tid < 192) {
        ga   = (unsigned long long)(const void*)((const char*)(pb + (size_t)n0 * 3) + tid * 16);
        ldso = lds_pts_off + tid * 16;
      } else {
        ga   = (unsigned long long)(const void*)((const char*)(q2 + n0) + (tid - 192) * 16);
        ldso = lds_p2_off + (tid - 192) * 16;
      }
      asm volatile("global_load_async_to_lds_b128 %0, %1, off"
                   :: "v"(ldso), "v"(ga) : "memory");
      asm volatile("s_wait_asynccnt 0" ::: "memory");
    }
    __syncthreads();

    // --- 16 sub-tiles of 16 points each ----------------------------------
#pragma unroll 1
    for (int sub = 0; sub < 16; ++sub) {
      const int nn = sub * 16 + lm;               // index within chunk
      const float* lp = &lds_pts[3 * nn];
      // B operand 4x16: lanes 0-15 -> K=0,1 at N=lane ; lanes 16-31 -> K=2,3
      v2f Bv;
      Bv.x = lo ? lp[0] : lp[2];
      Bv.y = lo ? lp[1] : 0.0f;

      v8f d0 = __builtin_amdgcn_wmma_f32_16x16x4_f32(
          false, A0, false, Bv, (short)0, zeroC, false, false);
      v8f d1 = __builtin_amdgcn_wmma_f32_16x16x4_f32(
          false, A1, false, Bv, (short)0, zeroC, false, false);

      // D layout: lane l holds column n=l%16, rows v + 8*(l>=16)
#pragma unroll
      for (int v = 0; v < 8; ++v) lds_cp[wv][rowhalf + v][lm] = d0[v];
#pragma unroll
      for (int v = 0; v < 8; ++v) lds_cp[wv][16 + rowhalf + v][lm] = d1[v];
      // same-wave LDS ops execute in order (DScnt in-order); just stop the
      // compiler from reordering across the store->load boundary.
      asm volatile("" ::: "memory");

#pragma unroll 1
      for (int j = 0; j < 16; ++j) {
        const float s = lds_p2[sub * 16 + j] - 2.0f * lds_cp[wv][l][j];
        const unsigned long long cand =
            ((unsigned long long)key_of(s) << 32) | (unsigned)(n0 + sub * 16 + j);
        if (cand < L[K - 1]) {
          unsigned long long v = cand;
#pragma unroll
          for (int q = 0; q < K; ++q) {
            const unsigned long long mn = (v < L[q]) ? v : L[q];
            const unsigned long long mx = (v < L[q]) ? L[q] : v;
            L[q] = mn;
            v = mx;
          }
        }
      }
      asm volatile("" ::: "memory");
    }
    __syncthreads();    // staged buffers reused next chunk
  }

  // finalize: mean of neighbors + mean distance
  float sx = 0.f, sy = 0.f, sz = 0.f, rs = 0.f;
#pragma unroll
  for (int q = 0; q < K; ++q) {
    const unsigned idx = (unsigned)(L[q] & 0xFFFFFFFFu);
    const float s = val_of((unsigned)(L[q] >> 32));
    const float* pp = pb + 3 * (size_t)idx;
    sx += pp[0]; sy += pp[1]; sz += pp[2];
    const float d2 = s + c2own;
    rs += sqrtf(fmaxf(d2, 0.0f));
  }
  const float inv = 1.0f / (float)K;
  const int m = mblk * 256 + wv * 32 + l;
  const size_t o = ((size_t)b * M + m);
  cout[o * 3 + 0] = sx * inv;
  cout[o * 3 + 1] = sy * inv;
  cout[o * 3 + 2] = sz * inv;
  rad[o] = rs * inv;
}

// ---------------------------------------------------------------------------
// Output assembly: centers [B,512,3] | sphere_points [B,512,128,3] | radius
// radius = (rad_fps_gathered + rad_sklnz2) * 1.5
// ---------------------------------------------------------------------------
__global__ __launch_bounds__(256)
void sphere_kernel(const float* __restrict__ cen,
                   const float* __restrict__ radA,
                   const float* __restrict__ radB,
                   const float* __restrict__ tmpl,
                   float* __restrict__ out) {
  const int total = NBATCH * NSKP * SPHERE_N;
  int t = blockIdx.x * blockDim.x + threadIdx.x;
  if (t >= total) return;
  const int s = t & (SPHERE_N - 1);
  const int m = (t >> 7) & (NSKP - 1);
  const int b = t >> 16;                      // 512*128 = 2^16
  const int cm = b * NSKP + m;

  const float r  = (radA[cm] + radB[cm]) * RAD_SCALE;
  const float cx = cen[3 * (size_t)cm + 0];
  const float cy = cen[3 * (size_t)cm + 1];
  const float cz = cen[3 * (size_t)cm + 2];

  const size_t SPH_OFF = (size_t)NBATCH * NSKP * 3;                       // 6144
  const size_t RAD_OFF = SPH_OFF + (size_t)NBATCH * NSKP * SPHERE_N * 3;  // 792576

  float* sph = out + SPH_OFF;
  sph[3 * (size_t)t + 0] = cx + r * tmpl[3 * s + 0];
  sph[3 * (size_t)t + 1] = cy + r * tmpl[3 * s + 1];
  sph[3 * (size_t)t + 2] = cz + r * tmpl[3 * s + 2];
  if (s == 0) {
    out[3 * (size_t)cm + 0] = cx;
    out[3 * (size_t)cm + 1] = cy;
    out[3 * (size_t)cm + 2] = cz;
    out[RAD_OFF + cm] = r;
  }
}

// ---------------------------------------------------------------------------
extern "C" void kernel_launch(void* const* d_in, const int* in_sizes, int n_in,
                              void* d_out, int out_size, void* d_ws, size_t ws_size,
                              hipStream_t stream) {
  (void)in_sizes; (void)n_in; (void)out_size; (void)ws_size;
  const float* xyz  = (const float*)d_in[0];  // [4,32768,3]
  const float* tmpl = (const float*)d_in[1];  // [128,3]
  float* ws = (float*)d_ws;

  // workspace layout (floats)
  float* cenA   = ws;                  // [4,2048,3]  = 24576
  float* cenB   = cenA + 24576;        // [4,2048,3]  = 24576
  float* rad1   = cenB + 24576;        // [4,2048]    =  8192
  float* p2a    = rad1 + 8192;         // [4,32768]   = 131072
  float* c5a    = p2a + 131072;        // [4,512,3]   =  6144
  float* c5b    = c5a + 6144;          // [4,512,3]   =  6144
  float* rad512 = c5b + 6144;          // [4,512]     =  2048
  float* rad2   = rad512 + 2048;       // [4,512]     =  2048
  float* p2b    = rad2 + 2048;         // [4,2048]    =  8192

  float* dout = (float*)d_out;

  // |p|^2 for the full cloud
  p2_kernel<<<(NBATCH * NPTS + 255) / 256, 256, 0, stream>>>(xyz, p2a, NBATCH * NPTS);

  // stage 1: FPS 32768 -> 2048 seeds (one block per batch)
  fps_kernel<NPTS / 1024, NSEED><<<NBATCH, 1024, 0, stream>>>(xyz, cenA, nullptr, nullptr);

  // stage 1: 2x skeletonization (K=32 vs full cloud)
  sklnz_kernel<KNN1, NSEED, NPTS><<<NBATCH * (NSEED / 256), 256, 0, stream>>>(
      cenA, xyz, p2a, cenB, rad1);
  sklnz_kernel<KNN1, NSEED, NPTS><<<NBATCH * (NSEED / 256), 256, 0, stream>>>(
      cenB, xyz, p2a, cenA, rad1);

  // stage 2: FPS 2048 -> 512 on the skeleton (also gathers radius)
  fps_kernel<NSEED / 1024, NSKP><<<NBATCH, 1024, 0, stream>>>(cenA, c5a, rad1, rad512);

  // |p|^2 for the intermediate skeleton
  p2_kernel<<<(NBATCH * NSEED + 255) / 256, 256, 0, stream>>>(cenA, p2b, NBATCH * NSEED);

  // stage 2: 2x skeletonization (K=2 vs 2048 skeleton points)
  sklnz_kernel<KNN2, NSKP, NSEED><<<NBATCH * (NSKP / 256), 256, 0, stream>>>(
      c5a, cenA, p2b, c5b, rad2);
  sklnz_kernel<KNN2, NSKP, NSEED><<<NBATCH * (NSKP / 256), 256, 0, stream>>>(
      c5b, cenA, p2b, c5a, rad2);

  // output: centers | sphere points | radius
  sphere_kernel<<<(NBATCH * NSKP * SPHERE_N + 255) / 256, 256, 0, stream>>>(
      c5a, rad512, rad2, tmpl, dout);
}